// SegmentTree_87995289960521
// MI455X (gfx1250) — compile-verified
//
#include <hip/hip_runtime.h>

// ---------------------------------------------------------------------------
// SegmentTree prefix-sum sampling for MI455X (gfx1250, wave32).
//
// tree heap layout in d_ws (float): tree[1] = root, node i children 2i,2i+1,
// internal nodes occupy [1, 2^24); leaves stay in d_in[0] (conceptually at
// heap [2^24, 2^25)).  Total ws use: 2^24 floats = 64 MB.
//
// Pass 1 (WMMA): level-1 pair sums leaf[2j]+leaf[2j+1] via
//   V_WMMA_F32_16X16X4_F32 with selector A:
//     A[0][k] = (k<2), A[1][k] = (k>=2), rows 2..15 = 0
//   B[k][n] = leaf[tile*64 + 4n + k]  (4x16 data tile)
//   => D[0][n] = pair(2n), D[1][n] = pair(2n+1); exact (a*1+b*1+0+0).
// Pass 2: generic parent = left+right per level (float2 child loads),
//         levels 2^22 .. 2^11 nodes.  All L2-resident (192 MB L2).
// Pass 3: single block finishes heap [1, 2048) in LDS.
// Pass 4: 65536 descent threads, 24 unrolled gather steps (L2 hits).
// ---------------------------------------------------------------------------

typedef __attribute__((ext_vector_type(2))) float v2f;
typedef __attribute__((ext_vector_type(8))) float v8f;

#define LEAVES   (1u << 24)
#define LVL1_OFF (1u << 23)   // heap offset of the first internal level

// ---- Pass 1: pair sums via WMMA (16x16x4 f32), one 64-leaf tile per step ---
__global__ void seg_build_l1_wmma(const float* __restrict__ leaf,
                                  float* __restrict__ tree, int ntiles) {
  const int lane  = threadIdx.x & 31;
  const int gwave = (blockIdx.x * (blockDim.x >> 5)) + (threadIdx.x >> 5);
  const int nwave = gridDim.x * (blockDim.x >> 5);

  // A (16x4) VGPR layout: lane m<16 holds {A[m][0], A[m][1]};
  //                       lane m+16 holds {A[m][2], A[m][3]}.
  // Want A[0][0..1]=1 (lane 0) and A[1][2..3]=1 (lane 17).
  const float asel = (lane == 0 || lane == 17) ? 1.0f : 0.0f;
  v2f a; a.x = asel; a.y = asel;

  for (int tile = gwave; tile < ntiles; tile += nwave) {
    const float* src = leaf + (size_t)tile * 64u;
    // B (4x16) layout: lane n<16 holds {B[0][n],B[1][n]} = leaf[4n..4n+1];
    //                  lane n+16 holds {B[2][n],B[3][n]} = leaf[4n+2..4n+3].
    const int n = lane & 15;
    v2f b = *(const v2f*)(src + 4 * n + 2 * (lane >> 4));

    v8f c = {};
    c = __builtin_amdgcn_wmma_f32_16x16x4_f32(
        /*neg_a=*/false, a, /*neg_b=*/false, b,
        /*c_mod=*/(short)0, c, /*reuse_a=*/false, /*reuse_b=*/false);

    // D layout: c[0] lane n<16 = D[0][n] = pair(2n); c[1] lane n = pair(2n+1).
    if (lane < 16) {
      v2f r; r.x = c[0]; r.y = c[1];
      *(v2f*)(tree + LVL1_OFF + (size_t)tile * 32u + 2u * (unsigned)lane) = r;
    }
  }
}

// ---- Pass 2: one heap level: tree[off+i] = tree[2(off+i)] + tree[2(off+i)+1]
__global__ void seg_build_level(float* __restrict__ tree, unsigned off) {
  unsigned i = blockIdx.x * blockDim.x + threadIdx.x;
  if (i < off) {
    v2f ch = *(const v2f*)(tree + 2u * (off + i));
    tree[off + i] = ch.x + ch.y;
  }
}

// ---- Pass 3: finish heap [1, 2048) in one block via LDS -------------------
__global__ void seg_build_top(float* __restrict__ tree) {
  __shared__ float sh[2048];
  const unsigned t = threadIdx.x;          // 1024 threads
  // level with 1024 nodes: children at heap [2048, 4096)
  v2f ch = *(const v2f*)(tree + 2048u + 2u * t);
  float s = ch.x + ch.y;
  sh[1024u + t] = s;
  tree[1024u + t] = s;
  __syncthreads();
  for (unsigned n = 512; n >= 1; n >>= 1) {
    if (t < n) {
      float v = sh[2u * (n + t)] + sh[2u * (n + t) + 1u];
      sh[n + t] = v;
      tree[n + t] = v;
    }
    __syncthreads();
  }
}

// ---- Pass 4: root-to-leaf binary descent ----------------------------------
__global__ void seg_descend(const float* __restrict__ leaf,
                            const float* __restrict__ tree,
                            const float* __restrict__ frac,
                            float* __restrict__ out_idx,
                            float* __restrict__ out_pri, int batch) {
  int i = blockIdx.x * blockDim.x + threadIdx.x;
  if (i >= batch) return;

  float v = frac[i] * tree[1];             // target in [0, root_sum)
  unsigned idx = 1u;
#pragma unroll
  for (int t = 0; t < 23; ++t) {           // internal levels (gathers hit L2)
    idx <<= 1;
    float l = tree[idx];
    bool go_right = l < v;
    v = go_right ? v - l : v;
    idx += (unsigned)go_right;
  }
  // final level: children are leaves; fetch both with one float2 gather
  idx <<= 1;                               // in [2^24, 2^25), even
  unsigned base = idx - LEAVES;            // even leaf index
  v2f lr = *(const v2f*)(leaf + base);
  bool go_right = lr.x < v;
  unsigned li = base + (unsigned)go_right;

  out_idx[i] = (float)li;                  // exact for li < 2^24
  out_pri[i] = go_right ? lr.y : lr.x;
}

// ---------------------------------------------------------------------------
extern "C" void kernel_launch(void* const* d_in, const int* in_sizes, int n_in,
                              void* d_out, int out_size, void* d_ws,
                              size_t ws_size, hipStream_t stream) {
  const float* leaf = (const float*)d_in[0];   // 2^24 leaf priorities
  const float* frac = (const float*)d_in[1];   // 65536 sample fractions
  float* out  = (float*)d_out;                 // [indices | priorities]
  float* tree = (float*)d_ws;                  // 2^24 floats (64 MB heap)

  const int ntiles = (int)(LEAVES / 64u);      // 262144 WMMA tiles
  seg_build_l1_wmma<<<1024, 256, 0, stream>>>(leaf, tree, ntiles);

  for (unsigned off = 1u << 22; off >= 2048u; off >>= 1)
    seg_build_level<<<(off + 255u) / 256u, 256, 0, stream>>>(tree, off);

  seg_build_top<<<1, 1024, 0, stream>>>(tree);

  const int batch = in_sizes[1];
  seg_descend<<<(batch + 255) / 256, 256, 0, stream>>>(
      leaf, tree, frac, out, out + batch, batch);
}